// GATNE_86260123173588
// MI455X (gfx1250) — compile-verified
//
#include <hip/hip_runtime.h>

#define NUM_B 4096
#define T_ 4
#define N_ 10
#define U_ 32
#define A_ 20
#define D_ 200
#define ROWS 16          // batch rows per block (one wave each)
#define K_ 128           // stacked K = T_*U_
#define NTILES 13        // ceil(200/16)

typedef __attribute__((ext_vector_type(2))) float v2f;
typedef __attribute__((ext_vector_type(8))) float v8f;

__device__ __forceinline__ float wave_sum(float v) {
#pragma unroll
  for (int off = 16; off > 0; off >>= 1)
    v += __shfl_xor(v, off, 32);
  return v;
}

__global__ __launch_bounds__(512, 2)
void gatne_fused_kernel(const int* __restrict__ train_inputs,
                        const int* __restrict__ train_types,
                        const int* __restrict__ node_neigh,
                        const float* __restrict__ node_embeddings,
                        const float* __restrict__ node_type_embeddings,
                        const float* __restrict__ trans_weights,     // (T,U,D) == (128,200)
                        const float* __restrict__ trans_w_s1,       // (T,U,A)
                        const float* __restrict__ trans_w_s2,       // (T,A,1)
                        float* __restrict__ out)                    // (B,D)
{
  __shared__ float s_nte[ROWS][T_][U_];      // per-row summed neighbor embeddings
  __shared__ float s_A[ROWS][132];           // stacked A matrix, padded stride (conflict-free)
  __shared__ float s_out[ROWS][NTILES * 16]; // GEMM tiles (208 wide; cols >= 200 are scratch)

  const int lane = threadIdx.x & 31;
  const int w    = threadIdx.x >> 5;              // wave id == batch row in block
  const int b    = blockIdx.x * ROWS + w;
  const int type = train_types[b];

  // ---------- Phase 1: coalesced gathers; lane == embedding dim u ----------
  float nte[T_];
#pragma unroll
  for (int t = 0; t < T_; ++t) {
    float acc = 0.f;
    const int* nidx = node_neigh + ((size_t)b * T_ + t) * N_;
#pragma unroll
    for (int n = 0; n < N_; ++n) {
      size_t idx = (size_t)nidx[n];
      acc += node_type_embeddings[(idx * T_ + t) * U_ + lane]; // 128B coalesced row
    }
    nte[t] = acc;
    s_nte[w][t][lane] = acc;
  }
  __syncthreads();

  // ---------- attention: h = tanh(nte @ s1), logits = h @ s2, softmax ----------
  const float* s1 = trans_w_s1 + (size_t)type * U_ * A_;
  const float* s2 = trans_w_s2 + (size_t)type * A_;
  float logit[T_];
#pragma unroll
  for (int t = 0; t < T_; ++t) {
    float acc = 0.f;
    if (lane < A_) {
#pragma unroll 8
      for (int u = 0; u < U_; ++u)
        acc += s_nte[w][t][u] * s1[u * A_ + lane];   // broadcast LDS read + cached weight
    }
    float h   = tanhf(acc);
    float val = (lane < A_) ? h * s2[lane] : 0.f;
    logit[t] = wave_sum(val);
  }
  float m  = fmaxf(fmaxf(logit[0], logit[1]), fmaxf(logit[2], logit[3]));
  float e0 = expf(logit[0] - m), e1 = expf(logit[1] - m);
  float e2 = expf(logit[2] - m), e3 = expf(logit[3] - m);
  float inv_s = 1.f / (e0 + e1 + e2 + e3);
  float agg = (e0 * nte[0] + e1 * nte[1] + e2 * nte[2] + e3 * nte[3]) * inv_s;

  // ---------- build stacked-A: row w = agg placed in its type's K-segment ----------
#pragma unroll
  for (int k = lane; k < K_; k += 32) s_A[w][k] = 0.f;
  s_A[w][type * U_ + lane] = agg;
  __syncthreads();

  // ---------- Phase 2: WMMA f32 GEMM  C(16x208) = A(16x128) @ W(128x200) ----------
  if (w < NTILES) {                       // wave-uniform -> EXEC all-1s for WMMA
    const int row  = lane & 15;
    const int kh   = (lane >> 4) << 1;    // 0 (lanes 0-15) or 2 (lanes 16-31)
    const int col  = w * 16 + (lane & 15);
    // Column-n of C depends only on column-n of B: out-of-range cols only pollute
    // scratch columns of s_out, so just clamp the ADDRESS (branchless, no exec mask).
    const int colc = col < D_ ? col : (D_ - 1);
    const float* __restrict__ Wb = trans_weights + colc;
    v8f c = {};
#pragma unroll 4
    for (int k0 = 0; k0 < K_; k0 += 4) {
      const int kk = k0 + kh;
      v2f a;
      a.x = s_A[row][kk];
      a.y = s_A[row][kk + 1];
      v2f bb;
      bb.x = Wb[(size_t)kk * D_];         // unconditional, 64B-coalesced per K-row
      bb.y = Wb[(size_t)(kk + 1) * D_];
      c = __builtin_amdgcn_wmma_f32_16x16x4_f32(false, a, false, bb,
                                                (short)0, c, false, false);
    }
    const int rbase = (lane >> 4) << 3;   // C layout: VGPR i -> M=i (lo half) / M=i+8 (hi half)
#pragma unroll
    for (int i = 0; i < 8; ++i)
      s_out[rbase + i][col] = c[i];
  }
  __syncthreads();

  // ---------- Phase 3: residual add + L2 normalize + store ----------
  const float* ne = node_embeddings + (size_t)train_inputs[b] * D_;
  float v[7];
  float ss = 0.f;
#pragma unroll
  for (int j = 0; j < 7; ++j) {
    int d = lane + j * 32;
    float x = 0.f;
    if (d < D_) { x = ne[d] + s_out[w][d]; ss += x * x; }
    v[j] = x;
  }
  ss = wave_sum(ss);
  float inv = 1.f / fmaxf(sqrtf(ss), 1e-12f);
#pragma unroll
  for (int j = 0; j < 7; ++j) {
    int d = lane + j * 32;
    if (d < D_) out[(size_t)b * D_ + d] = v[j] * inv;
  }
}

extern "C" void kernel_launch(void* const* d_in, const int* in_sizes, int n_in,
                              void* d_out, int out_size, void* d_ws, size_t ws_size,
                              hipStream_t stream) {
  (void)in_sizes; (void)n_in; (void)out_size; (void)d_ws; (void)ws_size;
  const int*   train_inputs         = (const int*)d_in[0];
  const int*   train_types          = (const int*)d_in[1];
  const int*   node_neigh           = (const int*)d_in[2];
  const float* node_embeddings      = (const float*)d_in[3];
  const float* node_type_embeddings = (const float*)d_in[4];
  const float* trans_weights        = (const float*)d_in[5];
  const float* trans_w_s1           = (const float*)d_in[6];
  const float* trans_w_s2           = (const float*)d_in[7];
  float* out = (float*)d_out;

  dim3 grid(NUM_B / ROWS);   // 256 blocks
  dim3 block(ROWS * 32);     // 512 threads = 16 waves
  gatne_fused_kernel<<<grid, block, 0, stream>>>(
      train_inputs, train_types, node_neigh, node_embeddings,
      node_type_embeddings, trans_weights, trans_w_s1, trans_w_s2, out);
}